// PretrainedCNN_53231824667088
// MI455X (gfx1250) — compile-verified
//
#include <hip/hip_runtime.h>

typedef __attribute__((ext_vector_type(2))) float v2f;
typedef __attribute__((ext_vector_type(8))) float v8f;

#define EPSF 1e-20f

__device__ __forceinline__ v8f wmma_f32_4(v2f a, v2f b, v8f c) {
  // D = A(16x4) * B(4x16) + C(16x16), f32, wave32
  return __builtin_amdgcn_wmma_f32_16x16x4_f32(false, a, false, b, (short)0, c, false, false);
}

// ---------------------------------------------------------------------------
// navg with 5x5 conv, Cin in {1,2}, Cout = 2.
//   ca   = conv(c)
//   xout = conv(x*c)/(ca+EPS) + bias
//   cout = ca / sum(w[oc])
// WMMA: M = 16 pixels, K = 4 (rows 0,1 = x*c channels, rows 2,3 = c channels),
// N cols 0,1 = x-conv outputs, cols 2,3 = c-conv outputs. 25 taps accumulated.
// A operands: channel-interleaved v2f LDS planes -> one ds_load_b64 per tap.
// B operands: pre-masked per-lane v2f table -> one ds_load_b64 per tap.
// Tile: 16 (W) x 32 (H), 256 threads = 8 waves, wave handles 4 rows.
// ---------------------------------------------------------------------------
__global__ __launch_bounds__(256)
void navg5_wmma(const float* __restrict__ xin, const float* __restrict__ cin,
                float* __restrict__ xout, float* __restrict__ cout,
                const float* __restrict__ w, const float* __restrict__ bias,
                int Cin, int H, int W)
{
  constexpr int TW = 16, TH = 32, HA = 2;
  constexpr int LW = TW + 2 * HA;   // 20
  constexpr int LH = TH + 2 * HA;   // 36
  __shared__ v2f s_xc[LH * LW];     // {xc0, xc1} per point
  __shared__ v2f s_c [LH * LW];     // {c0, c1} per point
  __shared__ v2f s_bw[25 * 32];     // pre-masked B: [tap][lane] = rows(K) of col N
  __shared__ float s_wsum[2];
  __shared__ float s_ep[8][64];     // per-wave epilogue staging: [pixel][col0..3]

  const int tid = threadIdx.x;
  const int b   = blockIdx.z;
  const int x0t = blockIdx.x * TW, y0t = blockIdx.y * TH;
  const size_t plane = (size_t)H * W;

  const float* xb = xin + (size_t)b * (size_t)Cin * plane;
  const float* cb = cin + (size_t)b * (size_t)Cin * plane;
  float* xo = xout + (size_t)b * 2 * plane;
  float* co = cout + (size_t)b * 2 * plane;

  // pre-masked B table: lane l supplies rows K=(l<16 ? 0,1 : 2,3) of column N=l&15.
  // cols 0,1 <- x-conv weights on xc rows; cols 2,3 <- c-conv weights on c rows.
  for (int i = tid; i < 25 * 32; i += 256) {
    int t = i >> 5, l = i & 31;
    int hi = l >> 4, m = l & 15;
    int oc0 = hi ? (m - 2) : m;
    bool use = (oc0 >= 0) && (oc0 < 2);
    int oc = use ? oc0 : 0;
    v2f bv;
    bv.x = use ? w[(oc * Cin + 0) * 25 + t] : 0.f;
    bv.y = (use && Cin > 1) ? w[(oc * Cin + 1) * 25 + t] : 0.f;
    s_bw[i] = bv;
  }
  // stage inputs: x*c and c planes, zero halo (conv 'same' zero padding)
  for (int i = tid; i < LH * LW; i += 256) {
    int r = i / LW, cc = i - r * LW;
    int y = y0t - HA + r, x = x0t - HA + cc;
    bool inb = (y >= 0) & (y < H) & (x >= 0) & (x < W);
    float xv0 = 0.f, cv0 = 0.f, xv1 = 0.f, cv1 = 0.f;
    if (inb) {
      size_t o = (size_t)y * W + x;
      xv0 = xb[o]; cv0 = cb[o];
      if (Cin > 1) { xv1 = xb[plane + o]; cv1 = cb[plane + o]; }
    }
    v2f t0; t0.x = xv0 * cv0; t0.y = xv1 * cv1;
    v2f t1; t1.x = cv0;       t1.y = cv1;
    s_xc[i] = t0;
    s_c [i] = t1;
  }
  __syncthreads();
  if (tid < 2) {
    // column 'tid' (hi=0 rows) of the B table holds the full (padded) kernel
    float s = 0.f;
    for (int t = 0; t < 25; ++t) { v2f v = s_bw[t * 32 + tid]; s += v.x + v.y; }
    s_wsum[tid] = s;
  }
  __syncthreads();

  const int wave = tid >> 5, lane = tid & 31;
  const int m = lane & 15, hi = lane >> 4;
  const v2f* pA = hi ? s_c : s_xc;    // hi==0: K=0,1 (x*c); hi==1: K=2,3 (c)
  const v2f* pB = s_bw + lane;
  const float bs0 = bias[0], bs1 = bias[1];

  for (int rr = 0; rr < 4; ++rr) {
    const int row = wave * 4 + rr;
    v8f acc = {};
#pragma unroll
    for (int t = 0; t < 25; ++t) {
      const int ty = t / 5, tx = t % 5;
      v2f a  = pA[(row + ty) * LW + (m + tx)];
      v2f bv = pB[t * 32];
      acc = wmma_f32_4(a, bv, acc);
    }
    // D layout: lane l, vgpr r -> M = r + 8*hi, N = l&15
    if (m < 4) {
#pragma unroll
      for (int r = 0; r < 8; ++r) s_ep[wave][(r + 8 * hi) * 4 + m] = acc[r];
    }
    __syncthreads();
    if (lane < 16) {
      const float xc0 = s_ep[wave][lane * 4 + 0];
      const float xc1 = s_ep[wave][lane * 4 + 1];
      const float cc0 = s_ep[wave][lane * 4 + 2];
      const float cc1 = s_ep[wave][lane * 4 + 3];
      const int gy = y0t + row, gx = x0t + lane;
      const size_t o = (size_t)gy * W + gx;
      xo[o]         = xc0 / (cc0 + EPSF) + bs0;
      xo[plane + o] = xc1 / (cc1 + EPSF) + bs1;
      co[o]         = cc0 / s_wsum[0];
      co[plane + o] = cc1 / s_wsum[1];
    }
    __syncthreads();
  }
}

// ---------------------------------------------------------------------------
// navg with 3x3 conv over a concat of two 2-channel tensors (Cin = 4).
// srcA -> input channels 0,1 ; srcB -> input channels 2,3.
// upA/upB: source is half resolution, nearest-upsampled on the fly.
// Two WMMA chains (x*c and c) with K = 4 = concat channels, shared B.
// final_mode: fuse trailing 1x1 navg (w4,b4), write 1-channel outputs.
// ---------------------------------------------------------------------------
__global__ __launch_bounds__(256)
void navg3cat_wmma(const float* __restrict__ xA, const float* __restrict__ cA, int upA,
                   const float* __restrict__ xB, const float* __restrict__ cB, int upB,
                   float* __restrict__ xout, float* __restrict__ cout,
                   const float* __restrict__ w, const float* __restrict__ bias,
                   const float* __restrict__ w4, const float* __restrict__ b4,
                   int final_mode, int H, int W)
{
  constexpr int TW = 16, TH = 32, HA = 1;
  constexpr int LW = TW + 2 * HA;   // 18
  constexpr int LH = TH + 2 * HA;   // 34
  __shared__ v2f s_xc[LH * LW][2];  // [point][half]: {xc(2h), xc(2h+1)}
  __shared__ v2f s_c [LH * LW][2];
  __shared__ v2f s_bw[9 * 32];      // pre-masked B: [tap][lane]
  __shared__ float s_wsum[2];
  __shared__ float s_ep[8][64];

  const int tid = threadIdx.x;
  const int b   = blockIdx.z;
  const int x0t = blockIdx.x * TW, y0t = blockIdx.y * TH;
  const size_t plane  = (size_t)H * W;
  const size_t planeS = plane / 4;         // half-res plane
  const int Ws = W >> 1;

  const float* xsrc[2] = { xA + (size_t)b * 2 * (upA ? planeS : plane),
                           xB + (size_t)b * 2 * (upB ? planeS : plane) };
  const float* csrc[2] = { cA + (size_t)b * 2 * (upA ? planeS : plane),
                           cB + (size_t)b * 2 * (upB ? planeS : plane) };
  const int ups[2] = { upA, upB };

  // pre-masked B: lane l supplies rows K=(l<16 ? 0,1 : 2,3) = ic (2*hi, 2*hi+1),
  // of column N = l&15; only cols 0,1 carry weights.
  for (int i = tid; i < 9 * 32; i += 256) {
    int t = i / 32, l = i & 31;
    int hi = l >> 4, m = l & 15;
    bool use = (m < 2);
    int oc = use ? m : 0;
    v2f bv;
    bv.x = use ? w[(oc * 4 + 2 * hi + 0) * 9 + t] : 0.f;
    bv.y = use ? w[(oc * 4 + 2 * hi + 1) * 9 + t] : 0.f;
    s_bw[i] = bv;
  }
  for (int i = tid; i < LH * LW; i += 256) {
    int r = i / LW, cc = i - r * LW;
    int y = y0t - HA + r, x = x0t - HA + cc;
    bool inb = (y >= 0) & (y < H) & (x >= 0) & (x < W);
#pragma unroll
    for (int half = 0; half < 2; ++half) {
      size_t pp  = ups[half] ? planeS : plane;
      size_t idx = ups[half] ? ((size_t)(y >> 1) * Ws + (x >> 1))
                             : ((size_t)y * W + x);
      float xv0 = 0.f, cv0 = 0.f, xv1 = 0.f, cv1 = 0.f;
      if (inb) {
        xv0 = xsrc[half][idx];      cv0 = csrc[half][idx];
        xv1 = xsrc[half][pp + idx]; cv1 = csrc[half][pp + idx];
      }
      v2f t0; t0.x = xv0 * cv0; t0.y = xv1 * cv1;
      v2f t1; t1.x = cv0;       t1.y = cv1;
      s_xc[i][half] = t0;
      s_c [i][half] = t1;
    }
  }
  __syncthreads();
  if (tid < 2) {
    // full kernel sum over all 4 input channels: cols tid (ic 0,1) + tid+16 (ic 2,3)
    float s = 0.f;
    for (int t = 0; t < 9; ++t) {
      v2f v0 = s_bw[t * 32 + tid];
      v2f v1 = s_bw[t * 32 + 16 + tid];
      s += v0.x + v0.y + v1.x + v1.y;
    }
    s_wsum[tid] = s;
  }
  __syncthreads();

  const int wave = tid >> 5, lane = tid & 31;
  const int m = lane & 15, hi = lane >> 4;
  const v2f* pB = s_bw + lane;
  const float bs0 = bias[0], bs1 = bias[1];
  const float w40 = w4[0], w41 = w4[1], bb4 = b4[0];

  for (int rr = 0; rr < 4; ++rr) {
    const int row = wave * 4 + rr;
    v8f accX = {}, accC = {};
#pragma unroll
    for (int t = 0; t < 9; ++t) {
      const int ty = t / 3, tx = t % 3;
      const int li = (row + ty) * LW + (m + tx);
      v2f aX = s_xc[li][hi];
      v2f aC = s_c [li][hi];
      v2f bv = pB[t * 32];
      accX = wmma_f32_4(aX, bv, accX);
      accC = wmma_f32_4(aC, bv, accC);
    }
    if (m < 2) {
#pragma unroll
      for (int r = 0; r < 8; ++r) {
        s_ep[wave][(r + 8 * hi) * 4 + m]     = accX[r];
        s_ep[wave][(r + 8 * hi) * 4 + 2 + m] = accC[r];
      }
    }
    __syncthreads();
    if (lane < 16) {
      const float xc0 = s_ep[wave][lane * 4 + 0];
      const float xc1 = s_ep[wave][lane * 4 + 1];
      const float cc0 = s_ep[wave][lane * 4 + 2];
      const float cc1 = s_ep[wave][lane * 4 + 3];
      const float X0 = xc0 / (cc0 + EPSF) + bs0;
      const float X1 = xc1 / (cc1 + EPSF) + bs1;
      const float C0 = cc0 / s_wsum[0];
      const float C1 = cc1 / s_wsum[1];
      const int gy = y0t + row, gx = x0t + lane;
      const size_t o = (size_t)gy * W + gx;
      if (!final_mode) {
        float* xo = xout + (size_t)b * 2 * plane;
        float* co = cout + (size_t)b * 2 * plane;
        xo[o] = X0; xo[plane + o] = X1;
        co[o] = C0; co[plane + o] = C1;
      } else {
        // fused 1x1 navg (w4, b4): 2 ch -> 1 ch
        const float ca2 = w40 * C0 + w41 * C1;
        const float xf  = (w40 * X0 * C0 + w41 * X1 * C1) / (ca2 + EPSF) + bb4;
        const float cf  = ca2 / (w40 + w41);
        xout[(size_t)b * plane + o] = xf;
        cout[(size_t)b * plane + o] = cf;
      }
    }
    __syncthreads();
  }
}

// ---------------------------------------------------------------------------
// 2x2 pool: argmax of c selects element; gather same element from x; c /= 4.
// Operates on [nimg, H, W] planes (nimg = B*2).
// ---------------------------------------------------------------------------
__global__ __launch_bounds__(256)
void pool_kernel(const float* __restrict__ x, const float* __restrict__ c,
                 float* __restrict__ xo, float* __restrict__ co,
                 int H, int W, int nimg)
{
  const int Ho = H / 2, Wo = W / 2;
  const long long total = (long long)nimg * Ho * Wo;
  long long idx = (long long)blockIdx.x * blockDim.x + threadIdx.x;
  if (idx >= total) return;
  int wo = (int)(idx % Wo);
  int ho = (int)((idx / Wo) % Ho);
  int pc = (int)(idx / ((long long)Wo * Ho));
  const float* cp = c + (size_t)pc * H * W;
  const float* xp = x + (size_t)pc * H * W;
  int y = 2 * ho, x0 = 2 * wo;
  float c00 = cp[(size_t)y * W + x0];
  float c01 = cp[(size_t)y * W + x0 + 1];
  float c10 = cp[(size_t)(y + 1) * W + x0];
  float c11 = cp[(size_t)(y + 1) * W + x0 + 1];
  int best = 0; float bc = c00;                       // argmax, first occurrence
  if (c01 > bc) { bc = c01; best = 1; }
  if (c10 > bc) { bc = c10; best = 2; }
  if (c11 > bc) { bc = c11; best = 3; }
  float xv = (best == 0) ? xp[(size_t)y * W + x0]
           : (best == 1) ? xp[(size_t)y * W + x0 + 1]
           : (best == 2) ? xp[(size_t)(y + 1) * W + x0]
                         : xp[(size_t)(y + 1) * W + x0 + 1];
  co[idx] = bc * 0.25f;
  xo[idx] = xv;
}

// ---------------------------------------------------------------------------
extern "C" void kernel_launch(void* const* d_in, const int* in_sizes, int n_in,
                              void* d_out, int out_size, void* d_ws, size_t ws_size,
                              hipStream_t stream)
{
  (void)in_sizes; (void)n_in; (void)out_size; (void)ws_size;
  const float* x0  = (const float*)d_in[0];
  const float* c0  = (const float*)d_in[1];
  const float* w1  = (const float*)d_in[2];
  const float* w2  = (const float*)d_in[3];
  const float* w3  = (const float*)d_in[4];
  const float* w4  = (const float*)d_in[5];
  const float* w34 = (const float*)d_in[6];
  const float* w23 = (const float*)d_in[7];
  const float* w12 = (const float*)d_in[8];
  const float* b1  = (const float*)d_in[9];
  const float* b2  = (const float*)d_in[10];
  const float* b3  = (const float*)d_in[11];
  const float* b4  = (const float*)d_in[12];
  const float* b34 = (const float*)d_in[13];
  const float* b23 = (const float*)d_in[14];
  const float* b12 = (const float*)d_in[15];

  const int B = 16, H = 512, W = 640;
  const size_t FULL = (size_t)B * 2 * H * W;   // floats per 2-ch full-res tensor
  const size_t HALF = FULL / 4, QUAR = FULL / 16, EIGH = FULL / 64;

  float* ws  = (float*)d_ws;
  float* F0x = ws;             float* F0c = F0x + FULL;
  float* F1x = F0c + FULL;     float* F1c = F1x + FULL;
  float* H0x = F1c + FULL;     float* H0c = H0x + HALF;
  float* H1x = H0c + HALF;     float* H1c = H1x + HALF;
  float* H2x = H1c + HALF;     float* H2c = H2x + HALF;
  float* Q0x = H2c + HALF;     float* Q0c = Q0x + QUAR;
  float* Q1x = Q0c + QUAR;     float* Q1c = Q1x + QUAR;
  float* E0x = Q1c + QUAR;     float* E0c = E0x + EIGH;
  float* E1x = E0c + EIGH;     float* E1c = E1x + EIGH;

  dim3 blk(256);
  auto g = [](int w, int h) { return dim3(w / 16, h / 32, 16); };

  // full-res encoder: w1, w2, w3  (x1,c1 land in F1)
  navg5_wmma<<<g(W, H), blk, 0, stream>>>(x0, c0,  F1x, F1c, w1, b1, 1, H, W);
  navg5_wmma<<<g(W, H), blk, 0, stream>>>(F1x, F1c, F0x, F0c, w2, b2, 2, H, W);
  navg5_wmma<<<g(W, H), blk, 0, stream>>>(F0x, F0c, F1x, F1c, w3, b3, 2, H, W);

  long long n;
  n = (long long)B * 2 * (H / 2) * (W / 2);
  pool_kernel<<<dim3((unsigned)((n + 255) / 256)), blk, 0, stream>>>(F1x, F1c, H0x, H0c, H, W, B * 2);
  navg5_wmma<<<g(W / 2, H / 2), blk, 0, stream>>>(H0x, H0c, H1x, H1c, w2, b2, 2, H / 2, W / 2);
  navg5_wmma<<<g(W / 2, H / 2), blk, 0, stream>>>(H1x, H1c, H2x, H2c, w3, b3, 2, H / 2, W / 2); // x2_ds

  n = (long long)B * 2 * (H / 4) * (W / 4);
  pool_kernel<<<dim3((unsigned)((n + 255) / 256)), blk, 0, stream>>>(H2x, H2c, Q0x, Q0c, H / 2, W / 2, B * 2);
  navg5_wmma<<<g(W / 4, H / 4), blk, 0, stream>>>(Q0x, Q0c, Q1x, Q1c, w2, b2, 2, H / 4, W / 4); // x3_ds

  n = (long long)B * 2 * (H / 8) * (W / 8);
  pool_kernel<<<dim3((unsigned)((n + 255) / 256)), blk, 0, stream>>>(Q1x, Q1c, E0x, E0c, H / 4, W / 4, B * 2);
  navg5_wmma<<<g(W / 8, H / 8), blk, 0, stream>>>(E0x, E0c, E1x, E1c, w2, b2, 2, H / 8, W / 8); // x4_ds

  // decoder: up2 fused via upA/upB indexing
  // x34 = navg(w34, cat(c3_ds, up2(c4)))  -> Q0
  navg3cat_wmma<<<g(W / 4, H / 4), blk, 0, stream>>>(Q1x, Q1c, 0, E1x, E1c, 1,
                                                     Q0x, Q0c, w34, b34, w4, b4, 0, H / 4, W / 4);
  // x23 = navg(w23, cat(c2_ds, up2(c34))) -> H0
  navg3cat_wmma<<<g(W / 2, H / 2), blk, 0, stream>>>(H2x, H2c, 0, Q0x, Q0c, 1,
                                                     H0x, H0c, w23, b23, w4, b4, 0, H / 2, W / 2);
  // final: navg(w12, cat(up2(c23), c1)) then fused 1x1 navg(w4) -> d_out
  float* OutX = (float*)d_out;
  float* OutC = OutX + (size_t)B * H * W;
  navg3cat_wmma<<<g(W, H), blk, 0, stream>>>(H0x, H0c, 1, F1x, F1c, 0,
                                             OutX, OutC, w12, b12, w4, b4, 1, H, W);
}